// PointGroup_7335804142301
// MI455X (gfx1250) — compile-verified
//
#include <hip/hip_runtime.h>

#define NPOINTS   500000
#define NC        32          // feature channels
#define NCLUSTER  256
#define FS        14
#define F3        (FS*FS*FS)  // 2744
#define CHUNK     16          // channels per LDS pass (2 passes)

// ---------- ordered-uint encoding for float atomic min/max ----------
__device__ __forceinline__ unsigned flipf(float f) {
    unsigned u = __float_as_uint(f);
    return u ^ ((unsigned)((int)u >> 31) | 0x80000000u);
}
__device__ __forceinline__ float unflipf(unsigned k) {
    unsigned mask = (k >> 31) ? 0x80000000u : 0xFFFFFFFFu;
    return __uint_as_float(k ^ mask);
}

// ---------- CDNA5 async global->LDS copy (ASYNCcnt-tracked) ----------
__device__ __forceinline__ void async_load_f32(unsigned lds_off, const float* gaddr) {
    asm volatile("global_load_async_to_lds_b32 %0, %1, off"
                 :: "v"(lds_off), "v"(gaddr) : "memory");
}
__device__ __forceinline__ void wait_async_le1() {
    asm volatile("s_wait_asynccnt 0x1" ::: "memory");
}
__device__ __forceinline__ void wait_async_le0() {
    asm volatile("s_wait_asynccnt 0x0" ::: "memory");
}

// ---------- workspace layout (bytes) ----------
#define WS_SUM   0        // float[NCLUSTER*4]  (sx,sy,sz,cnt)
#define WS_MEAN  4096     // float[NCLUSTER*3]
#define WS_MINK  7168     // u32  [NCLUSTER*3]
#define WS_MAXK  10240    // u32  [NCLUSTER*3]
#define WS_XF    13312    // float[NCLUSTER*4]  (scale, offx, offy, offz)
#define WS_SEG   17408    // int  [NCLUSTER+1]

// ================= kernel 1: init workspace =================
__global__ void pg_init(float* sum, float* mean, unsigned* mnk, unsigned* mxk, float* xf) {
    int t = threadIdx.x;
    for (int k = t; k < NCLUSTER*4; k += 256) { sum[k] = 0.f; xf[k] = 0.f; }
    for (int k = t; k < NCLUSTER*3; k += 256) {
        mean[k] = 0.f; mnk[k] = 0xFFFFFFFFu; mxk[k] = 0u;
    }
}

// ================= kernel 2: per-cluster coord sums (LDS privatized) =================
__global__ void pg_sums(const float* __restrict__ coords,
                        const int*   __restrict__ cid,
                        const int*   __restrict__ pid,
                        int S, float* gsum) {
    __shared__ float ls[NCLUSTER*4];
    int t = threadIdx.x;
    for (int k = t; k < NCLUSTER*4; k += 256) ls[k] = 0.f;
    __syncthreads();
    int base = blockIdx.x * 1024;
    #pragma unroll
    for (int i = 0; i < 4; ++i) {
        int s = base + t + i*256;
        if (s < S) {
            int c = cid[s], p = pid[s];
            atomicAdd(&ls[c*4+0], coords[p*3+0]);
            atomicAdd(&ls[c*4+1], coords[p*3+1]);
            atomicAdd(&ls[c*4+2], coords[p*3+2]);
            atomicAdd(&ls[c*4+3], 1.0f);
        }
    }
    __syncthreads();
    // sorted cids -> only ~2 clusters touched per block; flush conditionally
    if (ls[t*4+3] > 0.f) {
        atomicAdd(&gsum[t*4+0], ls[t*4+0]);
        atomicAdd(&gsum[t*4+1], ls[t*4+1]);
        atomicAdd(&gsum[t*4+2], ls[t*4+2]);
        atomicAdd(&gsum[t*4+3], ls[t*4+3]);
    }
}

// ================= kernel 3: mean + segment boundaries (binary search) =================
__global__ void pg_mean_seg(const int* __restrict__ cid, int S,
                            const float* __restrict__ gsum,
                            float* mean, int* seg) {
    int c = threadIdx.x;   // one thread per cluster
    int lo = 0, hi = S;    // lower_bound(cid, c)
    while (lo < hi) { int mid = (lo + hi) >> 1; if (cid[mid] < c) lo = mid + 1; else hi = mid; }
    seg[c] = lo;
    if (c == 0) seg[NCLUSTER] = S;
    float cnt = fmaxf(gsum[c*4+3], 1.0f);
    mean[c*3+0] = gsum[c*4+0] / cnt;
    mean[c*3+1] = gsum[c*4+1] / cnt;
    mean[c*3+2] = gsum[c*4+2] / cnt;
}

// ================= kernel 4: per-cluster min/max of recentered coords =================
__global__ void pg_minmax(const float* __restrict__ coords,
                          const int*   __restrict__ cid,
                          const int*   __restrict__ pid,
                          int S, const float* __restrict__ mean,
                          unsigned* gmn, unsigned* gmx) {
    __shared__ unsigned lmn[NCLUSTER*3], lmx[NCLUSTER*3];
    int t = threadIdx.x;
    for (int k = t; k < NCLUSTER*3; k += 256) { lmn[k] = 0xFFFFFFFFu; lmx[k] = 0u; }
    __syncthreads();
    int base = blockIdx.x * 1024;
    #pragma unroll
    for (int i = 0; i < 4; ++i) {
        int s = base + t + i*256;
        if (s < S) {
            int c = cid[s], p = pid[s];
            #pragma unroll
            for (int k = 0; k < 3; ++k) {
                float cc = coords[p*3+k] - mean[c*3+k];   // recenter like reference
                unsigned key = flipf(cc);
                atomicMin(&lmn[c*3+k], key);
                atomicMax(&lmx[c*3+k], key);
            }
        }
    }
    __syncthreads();
    if (lmn[t*3+0] != 0xFFFFFFFFu) {
        #pragma unroll
        for (int k = 0; k < 3; ++k) {
            atomicMin(&gmn[t*3+k], lmn[t*3+k]);
            atomicMax(&gmx[t*3+k], lmx[t*3+k]);
        }
    }
}

// ================= kernel 5: per-cluster scale & offset =================
__global__ void pg_scale(const unsigned* __restrict__ gmn,
                         const unsigned* __restrict__ gmx,
                         const int* __restrict__ seg, float* xf) {
    int c = threadIdx.x;
    if (seg[c] >= seg[c+1]) { xf[c*4]=0.f; xf[c*4+1]=0.f; xf[c*4+2]=0.f; xf[c*4+3]=0.f; return; }
    float cmin[3], cmax[3], r = 0.f;
    #pragma unroll
    for (int k = 0; k < 3; ++k) {
        cmin[k] = unflipf(gmn[c*3+k]);
        cmax[k] = unflipf(gmx[c*3+k]);
        r = fmaxf(r, (cmax[k] - cmin[k]) * (1.0f / FS));
    }
    float sc = 1.0f / r - 0.01f;       // r==0 -> +inf -> clamped to 50
    sc = fminf(sc, 50.0f);
    xf[c*4+0] = sc;
    xf[c*4+1] = -cmin[0] * sc;
    xf[c*4+2] = -cmin[1] * sc;
    xf[c*4+3] = -cmin[2] * sc;
}

// ================= kernel 6: voxelize + agg-max, one workgroup per cluster =================
// Dynamic LDS: vsum[F3*CHUNK] + vcnt[F3] = 186,592 B (CDNA5: 320 KB LDS per WGP).
// Feature gathers are software-pipelined through LDS with GLOBAL_LOAD_ASYNC_TO_LDS_B32:
// each wave double-buffers its 2 pairs' 64 B channel chunks (128 B per async instr),
// issuing batch b+1 before s_wait_asynccnt on batch b (async loads complete in order).
__global__ void pg_voxelize(const float* __restrict__ feats,
                            const float* __restrict__ coords,
                            const int*   __restrict__ pid,
                            const float* __restrict__ mean,
                            const float* __restrict__ xf,
                            const int*   __restrict__ seg,
                            float* __restrict__ out) {
    extern __shared__ float smem[];
    float* vsum = smem;               // [F3*CHUNK]
    float* vcnt = smem + F3*CHUNK;    // [F3]
    __shared__ float stage[8][2][32]; // [wave][buf][lane32] staged feature values
    __shared__ unsigned aggk[CHUNK];

    const int c      = blockIdx.x;
    const int t      = threadIdx.x;
    const int wave   = t >> 5;
    const int lane32 = t & 31;
    const int lane   = t & (CHUNK - 1);   // channel within chunk
    const int slot   = t >> 4;            // pair slot (16 pairs in flight per batch)
    const int start  = seg[c];
    const int end    = seg[c+1];

    const float sc = xf[c*4+0];
    const float ox = xf[c*4+1], oy = xf[c*4+2], oz = xf[c*4+3];
    const float mx = mean[c*3+0], my = mean[c*3+1], mz = mean[c*3+2];

    for (int chunk = 0; chunk < NC/CHUNK; ++chunk) {
        const int cbase = chunk * CHUNK;
        // zero LDS accumulators
        for (int k = t; k < F3*CHUNK; k += 256) vsum[k] = 0.f;
        for (int k = t; k < F3;       k += 256) vcnt[k] = 0.f;
        if (t < CHUNK) aggk[t] = 0u;
        __syncthreads();

        float amax = -__builtin_huge_valf();
        if (end > start) {
            const int nb = (end - start + 15) >> 4;   // block-uniform batch count

            // prologue: issue batch 0 (tail lanes clamped; exactly 1 async instr/wave)
            {
                int s0 = start + slot;
                int sc0 = (s0 < end) ? s0 : (end - 1);
                int p0 = pid[sc0];
                async_load_f32((unsigned)(uintptr_t)&stage[wave][0][lane32],
                               feats + (size_t)p0*NC + cbase + lane);
            }
            for (int b = 0; b < nb; ++b) {
                if (b + 1 < nb) {                      // issue next batch (uniform)
                    int s1 = start + slot + (b + 1)*16;
                    int sc1 = (s1 < end) ? s1 : (end - 1);
                    int p1 = pid[sc1];
                    async_load_f32((unsigned)(uintptr_t)&stage[wave][(b + 1) & 1][lane32],
                                   feats + (size_t)p1*NC + cbase + lane);
                    wait_async_le1();                  // batch b landed in LDS
                } else {
                    wait_async_le0();                  // last batch landed
                }
                int s = start + slot + b*16;
                if (s < end) {
                    float f = stage[wave][b & 1][lane32];
                    int p = pid[s];
                    float cx = (coords[p*3+0] - mx) * sc + ox;
                    float cy = (coords[p*3+1] - my) * sc + oy;
                    float cz = (coords[p*3+2] - mz) * sc + oz;
                    int vx = (int)fminf(fmaxf(floorf(cx), 0.f), (float)(FS-1));
                    int vy = (int)fminf(fmaxf(floorf(cy), 0.f), (float)(FS-1));
                    int vz = (int)fminf(fmaxf(floorf(cz), 0.f), (float)(FS-1));
                    int vloc = (vx*FS + vy)*FS + vz;
                    atomicAdd(&vsum[vloc*CHUNK + lane], f);   // ds_add_f32
                    if (lane == 0) atomicAdd(&vcnt[vloc], 1.0f);
                    amax = fmaxf(amax, f);
                }
            }
        }
        atomicMax(&aggk[lane], flipf(amax));
        __syncthreads();

        // write mean-pooled voxels (each output element written exactly once)
        for (int k = t; k < F3*CHUNK; k += 256) {
            int v = k >> 4, ch = k & (CHUNK - 1);
            float o = vsum[k] / fmaxf(vcnt[v], 1.0f);
            out[((size_t)c*F3 + v)*NC + cbase + ch] = o;
        }
        // write agg-max row (neg-inf -> 0 like reference)
        if (t < CHUNK) {
            float a = (end > start) ? unflipf(aggk[t]) : 0.0f;
            out[((size_t)NCLUSTER*F3 + c)*NC + cbase + t] = a;
        }
        __syncthreads();   // protect LDS before next chunk re-zeroes
    }
}

// ================= host launcher =================
extern "C" void kernel_launch(void* const* d_in, const int* in_sizes, int n_in,
                              void* d_out, int out_size, void* d_ws, size_t ws_size,
                              hipStream_t stream) {
    const float* feats  = (const float*)d_in[0];
    const float* coords = (const float*)d_in[1];
    const int*   cid    = (const int*)  d_in[2];
    const int*   pid    = (const int*)  d_in[3];
    const int    S      = in_sizes[2];
    float* out = (float*)d_out;

    char* ws = (char*)d_ws;
    float*    gsum = (float*)   (ws + WS_SUM);
    float*    mean = (float*)   (ws + WS_MEAN);
    unsigned* gmn  = (unsigned*)(ws + WS_MINK);
    unsigned* gmx  = (unsigned*)(ws + WS_MAXK);
    float*    xf   = (float*)   (ws + WS_XF);
    int*      seg  = (int*)     (ws + WS_SEG);

    const int statBlocks = (S + 1023) / 1024;
    const size_t voxSmem = (size_t)(F3*CHUNK + F3) * sizeof(float);  // 186,592 B

    pg_init    <<<1,          256, 0, stream>>>(gsum, mean, gmn, gmx, xf);
    pg_sums    <<<statBlocks, 256, 0, stream>>>(coords, cid, pid, S, gsum);
    pg_mean_seg<<<1,          256, 0, stream>>>(cid, S, gsum, mean, seg);
    pg_minmax  <<<statBlocks, 256, 0, stream>>>(coords, cid, pid, S, mean, gmn, gmx);
    pg_scale   <<<1,          256, 0, stream>>>(gmn, gmx, seg, xf);
    pg_voxelize<<<NCLUSTER,   256, voxSmem, stream>>>(feats, coords, pid, mean, xf, seg, out);
}